// ControlFieldModule_1606317768718
// MI455X (gfx1250) — compile-verified
//
#include <hip/hip_runtime.h>
#include <math.h>

typedef float v2f __attribute__((ext_vector_type(2)));
typedef float v8f __attribute__((ext_vector_type(8)));

#define D_MODEL 512
#define D_FIBER 32
#define D_CTRL  64
#define SEQ     4096
#define MOM     0.9f

__device__ __forceinline__ v8f wmma_f32(v2f a, v2f b, v8f c) {
    // V_WMMA_F32_16X16X4_F32 : D = A(16x4,f32) x B(4x16,f32) + C(16x16,f32)
    return __builtin_amdgcn_wmma_f32_16x16x4_f32(
        /*neg_a=*/false, a, /*neg_b=*/false, b,
        /*c_mod=*/(short)0, c, /*reuse_a=*/false, /*reuse_b=*/false);
}

// One wave per 32-token block (2 M-tiles): each B fragment loaded once and
// reused across both M-tiles, halving weight cache traffic vs 1 tile/wave.
// Fuses: fiber GEMM (+bias, store), concat GEMM, GELU, dot(w2), softplus+clip.
__global__ __launch_bounds__(32) void cf_pertoken(
    const float* __restrict__ hidden,
    const float* __restrict__ w_fiber, const float* __restrict__ b_fiber,
    const float* __restrict__ w1, const float* __restrict__ b1,
    const float* __restrict__ w2, const float* __restrict__ b2,
    float* __restrict__ out_fiber, float* __restrict__ out_delta)
{
    __shared__ float lds_fiber[32 * D_FIBER];   // 4 KB: C-layout -> A-layout bridge

    const int lane = threadIdx.x;
    const int half = lane >> 4;        // selects K-pair within the k-step of 4
    const int lm   = lane & 15;        // M for A-loads, N for B-loads/C-layout
    const long token_base = (long)blockIdx.x * 32;
    const float* Arow0 = hidden + (token_base +      lm) * D_MODEL;
    const float* Arow1 = hidden + (token_base + 16 + lm) * D_MODEL;

    // ---------- GEMM 1: fiber[32x32] = hidden[32x512] @ w_fiber[512x32] ----------
    v8f cf[2][2] = {{{}, {}}, {{}, {}}};
    for (int k0 = 0; k0 < D_MODEL; k0 += 4) {
        const int kk = k0 + half * 2;
        v2f b0, b1f;
        b0.x  = w_fiber[(kk    ) * D_FIBER + lm];
        b0.y  = w_fiber[(kk + 1) * D_FIBER + lm];
        b1f.x = w_fiber[(kk    ) * D_FIBER + 16 + lm];
        b1f.y = w_fiber[(kk + 1) * D_FIBER + 16 + lm];
        v2f a0 = *(const v2f*)(Arow0 + kk);
        v2f a1 = *(const v2f*)(Arow1 + kk);
        cf[0][0] = wmma_f32(a0, b0,  cf[0][0]);
        cf[0][1] = wmma_f32(a0, b1f, cf[0][1]);
        cf[1][0] = wmma_f32(a1, b0,  cf[1][0]);
        cf[1][1] = wmma_f32(a1, b1f, cf[1][1]);
    }
    const float bf0 = b_fiber[lm], bf1 = b_fiber[16 + lm];
    #pragma unroll
    for (int m = 0; m < 2; ++m) {
        #pragma unroll
        for (int r = 0; r < 8; ++r) {
            // C-layout: VGPR r, lanes 0-15 -> M=r ; lanes 16-31 -> M=r+8
            const int   row = m * 16 + r + 8 * half;
            const float f0  = cf[m][0][r] + bf0;
            const float f1  = cf[m][1][r] + bf1;
            lds_fiber[row * D_FIBER + lm]      = f0;
            lds_fiber[row * D_FIBER + 16 + lm] = f1;
            const long tok = token_base + row;
            out_fiber[tok * D_FIBER + lm]      = f0;
            out_fiber[tok * D_FIBER + 16 + lm] = f1;
        }
    }
    __syncthreads();

    // ---------- GEMM 2: pre[32x64] = [hidden|fiber][32x544] @ w1[544x64] ----------
    v8f ch[2][4] = {{{}, {}, {}, {}}, {{}, {}, {}, {}}};
    for (int k0 = 0; k0 < D_MODEL; k0 += 4) {
        const int kk = k0 + half * 2;
        v2f a0 = *(const v2f*)(Arow0 + kk);
        v2f a1 = *(const v2f*)(Arow1 + kk);
        #pragma unroll
        for (int t = 0; t < 4; ++t) {
            v2f b;
            b.x = w1[(kk    ) * D_CTRL + t * 16 + lm];
            b.y = w1[(kk + 1) * D_CTRL + t * 16 + lm];
            ch[0][t] = wmma_f32(a0, b, ch[0][t]);
            ch[1][t] = wmma_f32(a1, b, ch[1][t]);
        }
    }
    // concat part: K = 512..543 comes from the fiber tiles staged in LDS
    for (int k0 = 0; k0 < D_FIBER; k0 += 4) {
        const int kk = k0 + half * 2;
        v2f a0 = *(const v2f*)(&lds_fiber[(     lm) * D_FIBER + kk]);  // ds_load_b64
        v2f a1 = *(const v2f*)(&lds_fiber[(16 + lm) * D_FIBER + kk]);
        #pragma unroll
        for (int t = 0; t < 4; ++t) {
            v2f b;
            b.x = w1[(D_MODEL + kk    ) * D_CTRL + t * 16 + lm];
            b.y = w1[(D_MODEL + kk + 1) * D_CTRL + t * 16 + lm];
            ch[0][t] = wmma_f32(a0, b, ch[0][t]);
            ch[1][t] = wmma_f32(a1, b, ch[1][t]);
        }
    }

    // ---------- bias + exact GELU + dot(w2) + softplus + clip ----------
    float w2v[4], b1v[4];
    #pragma unroll
    for (int t = 0; t < 4; ++t) { w2v[t] = w2[t * 16 + lm]; b1v[t] = b1[t * 16 + lm]; }
    const float b2s = b2[0];
    #pragma unroll
    for (int m = 0; m < 2; ++m) {
        #pragma unroll
        for (int r = 0; r < 8; ++r) {
            float partial = 0.f;
            #pragma unroll
            for (int t = 0; t < 4; ++t) {
                const float x = ch[m][t][r] + b1v[t];
                const float g = 0.5f * x * (1.0f + erff(x * 0.70710678118654752f));
                partial += g * w2v[t];
            }
            // sum over N (16 lanes within each half; offsets 1..8 never cross halves)
            partial += __shfl_xor(partial, 1, 32);
            partial += __shfl_xor(partial, 2, 32);
            partial += __shfl_xor(partial, 4, 32);
            partial += __shfl_xor(partial, 8, 32);
            const float x  = partial + b2s;
            float sp = (x > 20.f) ? x : log1pf(expf(x));        // softplus
            sp = fminf(fmaxf(sp, 0.f), 1.f);                    // clip [0,1]
            if (lm == 0) out_delta[token_base + m * 16 + r + 8 * half] = sp;
        }
    }
}

// 8 blocks x 1 wave: EMA scan field_t = m*field_{t-1} + (1-m)*delta_t via
// intra-wave affine scan, sequential carry across 128 chunks of 32.
__global__ __launch_bounds__(32) void cf_scan(
    const float* __restrict__ delta, const float* __restrict__ lambda_gate,
    float* __restrict__ gate, float* __restrict__ field_out, int seq)
{
    const int lane = threadIdx.x;
    const long base = (long)blockIdx.x * seq;
    const float lam = lambda_gate[0];
    float carry = 0.f;
    for (int c = 0; c < seq; c += 32) {
        const float dh = delta[base + c + lane];
        float A = MOM;
        float B = (1.f - MOM) * dh;
        #pragma unroll
        for (int off = 1; off < 32; off <<= 1) {
            const float Au = __shfl_up(A, off, 32);
            const float Bu = __shfl_up(B, off, 32);
            if (lane >= off) { B = A * Bu + B; A = A * Au; }  // combine(up, cur)
        }
        const float field = A * carry + B;        // apply incoming carry
        carry = __shfl(field, 31, 32);            // last element -> next chunk
        const float fc = fminf(fmaxf(field, 0.f), 10.f);
        gate[base + c + lane]      = 1.0f / (1.0f + expf(lam * fc));  // sigmoid(-lam*f)
        field_out[base + c + lane] = fc;
    }
}

extern "C" void kernel_launch(void* const* d_in, const int* in_sizes, int n_in,
                              void* d_out, int out_size, void* d_ws, size_t ws_size,
                              hipStream_t stream) {
    const float* hidden  = (const float*)d_in[0];
    const float* w_fiber = (const float*)d_in[1];
    const float* b_fiber = (const float*)d_in[2];
    const float* w1      = (const float*)d_in[3];
    const float* b1      = (const float*)d_in[4];
    const float* w2      = (const float*)d_in[5];
    const float* b2      = (const float*)d_in[6];
    const float* lam     = (const float*)d_in[7];

    const int n_tokens = in_sizes[0] / D_MODEL;     // B*S = 32768
    const int rows     = n_tokens / SEQ;            // B = 8

    // d_out = [gate | field | delta_h | fiber], flat in return order
    float* out_gate  = (float*)d_out;
    float* out_field = out_gate  + n_tokens;
    float* out_delta = out_field + n_tokens;
    float* out_fiber = out_delta + n_tokens;

    cf_pertoken<<<n_tokens / 32, 32, 0, stream>>>(
        hidden, w_fiber, b_fiber, w1, b1, w2, b2, out_fiber, out_delta);
    cf_scan<<<rows, 32, 0, stream>>>(out_delta, lam, out_gate, out_field, SEQ);
}